// DeepGEMMGroupedLinear_83133386982049
// MI455X (gfx1250) — compile-verified
//
#include <hip/hip_runtime.h>

typedef __attribute__((ext_vector_type(2))) float v2f;
typedef __attribute__((ext_vector_type(8))) float v8f;

#define NUM_GROUPS 8
#define BM 128
#define BN 128
#define BK 16
#define LDT 20   // padded LDS row stride (floats): conflict-free float2 fragment reads

__global__ __launch_bounds__(256, 2)
void grouped_gemm_wmma_f32(const float* __restrict__ x,
                           const float* __restrict__ w,
                           const float* __restrict__ bias,
                           const int* __restrict__ gidx,
                           float* __restrict__ out,
                           int N, int K, int DOUT) {
  __shared__ float lA[2][BM * LDT];
  __shared__ float lB[2][BN * LDT];

  const int tid  = threadIdx.x;
  const int lane = tid & 31;
  const int wave = tid >> 5;
  const int wm   = wave >> 2;   // 0..1 : 64-row slab
  const int wn   = wave & 3;    // 0..3 : 32-col slab
  const int hl   = lane >> 4;   // lane half (K-pair / M+8 select)
  const int lrow = lane & 15;

  const int row0 = blockIdx.y * BM;
  const int col0 = blockIdx.x * BN;

  // per-thread staging geometry: 2 x 16B per tile per stage
  const int q0 = tid,       r0 = q0 >> 2, c0 = (q0 & 3) << 2;
  const int q1 = tid + 256, r1 = q1 >> 2, c1 = (q1 & 3) << 2;
  const unsigned va0 = (unsigned)((r0 * K + c0) * 4);  // global byte offset (row-space)
  const unsigned va1 = (unsigned)((r1 * K + c1) * 4);

  // sorted indices => contiguous group span for this row block
  const int g_first = gidx[row0];
  const int g_last  = gidx[row0 + BM - 1];

  for (int g = g_first; g <= g_last; ++g) {
    const float* __restrict__ wg = w + (size_t)g * DOUT * K;

    // async global->LDS stage issue (4 x b128 per thread)
    auto issue = [&](int kt, int buf) {
      const float* sa = x  + (size_t)row0 * K + kt;
      const float* sb = wg + (size_t)col0 * K + kt;
      unsigned la0 = (unsigned)(unsigned long long)(&lA[buf][r0 * LDT + c0]);
      unsigned la1 = (unsigned)(unsigned long long)(&lA[buf][r1 * LDT + c1]);
      unsigned lb0 = (unsigned)(unsigned long long)(&lB[buf][r0 * LDT + c0]);
      unsigned lb1 = (unsigned)(unsigned long long)(&lB[buf][r1 * LDT + c1]);
      asm volatile("global_load_async_to_lds_b128 %0, %1, %2"
                   :: "v"(la0), "v"(va0), "s"(sa) : "memory");
      asm volatile("global_load_async_to_lds_b128 %0, %1, %2"
                   :: "v"(la1), "v"(va1), "s"(sa) : "memory");
      asm volatile("global_load_async_to_lds_b128 %0, %1, %2"
                   :: "v"(lb0), "v"(va0), "s"(sb) : "memory");
      asm volatile("global_load_async_to_lds_b128 %0, %1, %2"
                   :: "v"(lb1), "v"(va1), "s"(sb) : "memory");
    };

    // init accumulators with bias (col-only dependent)
    v8f acc[4][2];
#pragma unroll
    for (int j = 0; j < 2; ++j) {
      float bv = bias[(size_t)g * DOUT + col0 + wn * 32 + j * 16 + lrow];
#pragma unroll
      for (int i = 0; i < 4; ++i)
#pragma unroll
        for (int e = 0; e < 8; ++e) acc[i][j][e] = bv;
    }

    const int nk = K / BK;
    issue(0, 0);
    for (int s = 0; s < nk; ++s) {
      const int buf = s & 1;
      if (s + 1 < nk) {
        issue((s + 1) * BK, buf ^ 1);
        // in-order async completion: <=4 outstanding == stage s fully landed
        asm volatile("s_wait_asynccnt 0x4" ::: "memory");
      } else {
        asm volatile("s_wait_asynccnt 0x0" ::: "memory");
      }
      __syncthreads();   // all waves' async writes visible

      // 4 micro-steps of K=4, 8 WMMAs each
#pragma unroll
      for (int kk = 0; kk < BK; kk += 4) {
        v2f af[4], bf[2];
#pragma unroll
        for (int i = 0; i < 4; ++i) {
          int m = wm * 64 + i * 16 + lrow;
          af[i] = *(const v2f*)(&lA[buf][m * LDT + kk + 2 * hl]);
        }
#pragma unroll
        for (int j = 0; j < 2; ++j) {
          int n = wn * 32 + j * 16 + lrow;
          bf[j] = *(const v2f*)(&lB[buf][n * LDT + kk + 2 * hl]);
        }
#pragma unroll
        for (int i = 0; i < 4; ++i)
#pragma unroll
          for (int j = 0; j < 2; ++j)
            acc[i][j] = __builtin_amdgcn_wmma_f32_16x16x4_f32(
                false, af[i], false, bf[j], (short)0, acc[i][j], false, false);
      }
      __syncthreads();   // reads done before next issue overwrites this buffer
    }

    // masked epilogue: only rows belonging to group g (nontemporal, write-once)
#pragma unroll
    for (int i = 0; i < 4; ++i) {
      int mbase = row0 + wm * 64 + i * 16 + hl * 8;
#pragma unroll
      for (int e = 0; e < 8; ++e) {
        int r = mbase + e;
        if (gidx[r] == g) {
#pragma unroll
          for (int j = 0; j < 2; ++j) {
            int c = col0 + wn * 32 + j * 16 + lrow;
            __builtin_nontemporal_store(acc[i][j][e], &out[(size_t)r * DOUT + c]);
          }
        }
      }
    }
  }
}

extern "C" void kernel_launch(void* const* d_in, const int* in_sizes, int n_in,
                              void* d_out, int out_size, void* d_ws, size_t ws_size,
                              hipStream_t stream) {
  (void)d_ws; (void)ws_size; (void)n_in; (void)out_size;
  const float* x    = (const float*)d_in[0];
  const float* w    = (const float*)d_in[1];
  const float* b    = (const float*)d_in[2];
  const int*   gidx = (const int*)d_in[3];
  float* out = (float*)d_out;

  const int N    = in_sizes[3];                 // 16384
  const int DOUT = in_sizes[2] / NUM_GROUPS;    // 2048
  const int K    = in_sizes[0] / N;             // 2048

  dim3 grid(DOUT / BN, N / BM);                 // (16, 128)
  grouped_gemm_wmma_f32<<<grid, 256, 0, stream>>>(x, w, b, gidx, out, N, K, DOUT);
}